// GCN_55602646614062
// MI455X (gfx1250) — compile-verified
//
#include <hip/hip_runtime.h>

// CDNA5 / gfx1250 GCN 2-layer forward.
// GEMMs use V_WMMA_F32_16X16X4_F32 (fp32-exact vs reference). Edge scatter is
// L2-resident (xw slab = 25.6MB << 192MB L2) and uses relaxed agent-scope
// __hip_atomic_fetch_add to guarantee single-instruction global_atomic_add_f32,
// with b128/b64 vector gathers of the source rows.

typedef __attribute__((ext_vector_type(2))) float v2f;
typedef __attribute__((ext_vector_type(4))) float v4f;
typedef __attribute__((ext_vector_type(8))) float v8f;

#define SELF_LOOP_W 2.0f

__device__ __forceinline__ void atomic_add_f32(float* p, float v) {
  __hip_atomic_fetch_add(p, v, __ATOMIC_RELAXED, __HIP_MEMORY_SCOPE_AGENT);
}

// ---------------- degree / normalization ----------------

__global__ void k_init_deg(float* __restrict__ deg, int n) {
  int i = blockIdx.x * blockDim.x + threadIdx.x;
  if (i < n) deg[i] = SELF_LOOP_W;
}

__global__ void k_accum_deg(float* __restrict__ deg, const int* __restrict__ dst,
                            const float* __restrict__ ew, int e) {
  int i = blockIdx.x * blockDim.x + threadIdx.x;
  if (i < e) atomic_add_f32(&deg[dst[i]], ew[i]);
}

__global__ void k_rsqrt_inplace(float* __restrict__ d, int n) {
  int i = blockIdx.x * blockDim.x + threadIdx.x;
  if (i < n) d[i] = rsqrtf(d[i]);
}

__global__ void k_edge_norm(float* __restrict__ norm, const float* __restrict__ dinv,
                            const int* __restrict__ src, const int* __restrict__ dst,
                            const float* __restrict__ ew, int e) {
  int i = blockIdx.x * blockDim.x + threadIdx.x;
  if (i < e) norm[i] = dinv[src[i]] * ew[i] * dinv[dst[i]];
}

// ---------------- fp32 WMMA GEMM: Y[n x M] = X[n x K] @ W[K x M] ----------------
// grid.x = ceil(n/16); block = (32, M/16). One wave per 16x16 output tile.
// A frag (16x4 f32): lane L<16 holds A[M=L][K=k,k+1] in v0,v1; lanes 16-31 hold K=k+2,k+3.
// B frag (4x16 f32): lane L<16 holds B[K=k][N=L] (v0), B[K=k+1][N=L] (v1); lanes 16-31: K=k+2,k+3.
// C/D (16x16 f32): VGPR i: lanes 0-15 -> row i, lanes 16-31 -> row i+8, col = lane&15.
__global__ void k_gemm_wmma_f32(const float* __restrict__ X, const float* __restrict__ W,
                                float* __restrict__ Y, int n, int K, int M) {
  const int lane  = threadIdx.x & 31;
  const int col0  = threadIdx.y * 16;
  const int rb    = blockIdx.x * 16;
  const int m15   = lane & 15;
  const int khalf = (lane >> 4) * 2;          // 0 or 2

  int arow = rb + m15;
  if (arow >= n) arow = n - 1;                // clamp loads; EXEC stays all-ones for WMMA
  const float* xrow = X + (long)arow * K;
  const int    bcol = col0 + m15;

  v8f acc = {};
#if __has_builtin(__builtin_amdgcn_wmma_f32_16x16x4_f32)
  for (int k = 0; k < K; k += 4) {
    v2f a = *(const v2f*)(xrow + k + khalf);  // contiguous pair, 8B aligned
    v2f b;
    b.x = W[(long)(k + khalf + 0) * M + bcol];
    b.y = W[(long)(k + khalf + 1) * M + bcol];
    acc = __builtin_amdgcn_wmma_f32_16x16x4_f32(
        /*neg_a=*/false, a, /*neg_b=*/false, b,
        /*c_mod=*/(short)0, acc, /*reuse_a=*/false, /*reuse_b=*/false);
  }
#else
#warning "gfx1250: __builtin_amdgcn_wmma_f32_16x16x4_f32 not available; scalar fallback in use"
  const int mbase0 = (lane >> 4) * 8;
  for (int i = 0; i < 8; ++i) {
    int r = rb + mbase0 + i; if (r >= n) r = n - 1;
    float s = 0.f;
    for (int k = 0; k < K; ++k) s += X[(long)r * K + k] * W[(long)k * M + bcol];
    acc[i] = s;
  }
#endif

  const int mbase = (lane >> 4) * 8;
  #pragma unroll
  for (int i = 0; i < 8; ++i) {
    int r = rb + mbase + i;
    if (r < n) Y[(long)r * M + col0 + m15] = acc[i];
  }
}

// ---------------- self-loop init: agg = xw * (2 * dinv^2)  (M = 1<<shift) ----------------

__global__ void k_self_init(float* __restrict__ agg, const float* __restrict__ xw,
                            const float* __restrict__ dinv, long total, int shift) {
  long i = (long)blockIdx.x * blockDim.x + threadIdx.x;
  if (i < total) {
    int r = (int)(i >> shift);
    float di = dinv[r];
    agg[i] = xw[i] * (SELF_LOOP_W * di * di);
  }
}

// ---------------- edge scatter: agg[dst] += xw[src] * norm ----------------
// One wave per edge; V contiguous columns per lane (V = M/32) so the source-row
// gather is a single global_load_b128 (V=4) / b64 (V=2) per lane.

template <int V>
__global__ void k_scatter_vec(float* __restrict__ agg, const float* __restrict__ xw,
                              const float* __restrict__ norm, const int* __restrict__ src,
                              const int* __restrict__ dst, int e) {
  const int M    = V * 32;
  int gw   = (int)(((long)blockIdx.x * blockDim.x + threadIdx.x) >> 5);
  int lane = threadIdx.x & 31;
  if (gw >= e) return;                 // uniform per wave
  int s = src[gw], d = dst[gw];
  float w = norm[gw];
  const float* xs = xw  + (long)s * M + lane * V;
  float*       ad = agg + (long)d * M + lane * V;
  float v[V];
  if constexpr (V == 4) {
    v4f t = *(const v4f*)xs;           // 16B-aligned: rows are 512B-aligned
    v[0] = t.x; v[1] = t.y; v[2] = t.z; v[3] = t.w;
  } else if constexpr (V == 2) {
    v2f t = *(const v2f*)xs;
    v[0] = t.x; v[1] = t.y;
  } else {
    #pragma unroll
    for (int i = 0; i < V; ++i) v[i] = xs[i];
  }
  #pragma unroll
  for (int i = 0; i < V; ++i) atomic_add_f32(&ad[i], v[i] * w);
}

__global__ void k_scatter_generic(float* __restrict__ agg, const float* __restrict__ xw,
                                  const float* __restrict__ norm, const int* __restrict__ src,
                                  const int* __restrict__ dst, int e, int M) {
  int gw   = (int)(((long)blockIdx.x * blockDim.x + threadIdx.x) >> 5);
  int lane = threadIdx.x & 31;
  if (gw >= e) return;
  int s = src[gw], d = dst[gw];
  float w = norm[gw];
  const float* xs = xw + (long)s * M;
  float*       ad = agg + (long)d * M;
  for (int c = lane; c < M; c += 32) atomic_add_f32(&ad[c], xs[c] * w);
}

// ---------------- bias + relu (in place), M = 1<<shift ----------------

__global__ void k_bias_relu(float* __restrict__ h, const float* __restrict__ bias,
                            long total, int shift) {
  long i = (long)blockIdx.x * blockDim.x + threadIdx.x;
  if (i < total) {
    int c = (int)(i & ((1 << shift) - 1));
    float v = h[i] + bias[c];
    h[i] = v > 0.f ? v : 0.f;
  }
}

// ---------------- final fc: out[r] = h[r,:] . Wfc + bfc ----------------

__global__ void k_fc(float* __restrict__ out, const float* __restrict__ h,
                     const float* __restrict__ w, const float* __restrict__ b,
                     int n, int M) {
  int r = blockIdx.x * blockDim.x + threadIdx.x;
  if (r < n) {
    const float* hr = h + (long)r * M;
    float s = 0.f;
    for (int c = 0; c < M; ++c) s += hr[c] * w[c];
    out[r] = s + b[0];
  }
}

// ---------------- host orchestration ----------------

static inline int ilog2i(int v) { int s = 0; while ((1 << s) < v) ++s; return s; }

extern "C" void kernel_launch(void* const* d_in, const int* in_sizes, int n_in,
                              void* d_out, int out_size, void* d_ws, size_t ws_size,
                              hipStream_t stream) {
  const float* x   = (const float*)d_in[0];
  const int*   ei  = (const int*)d_in[1];   // harness contract: integer -> int32
  const float* ew  = (const float*)d_in[2];
  const float* W1  = (const float*)d_in[3];
  const float* b1  = (const float*)d_in[4];
  const float* W2  = (const float*)d_in[5];
  const float* b2  = (const float*)d_in[6];
  const float* Wfc = (const float*)d_in[7];
  const float* bfc = (const float*)d_in[8];
  float* out = (float*)d_out;

  const int E  = in_sizes[2];           // edge_weight count
  const int H1 = in_sizes[4];           // 128
  const int H2 = in_sizes[6];           // 64
  const int F  = in_sizes[3] / H1;      // 128
  const int N  = in_sizes[0] / F;       // 50000
  const int sh1 = ilog2i(H1), sh2 = ilog2i(H2);

  const int* src = ei;
  const int* dst = ei + E;

  // workspace carve-out (256B-aligned slabs)
  size_t off = 0;
  auto carve = [&](size_t bytes) -> float* {
    float* p = (float*)((char*)d_ws + off);
    off += (bytes + 255) & ~(size_t)255;
    return p;
  };
  float* dinv = carve((size_t)N * 4);        // deg, then rsqrt'd in place
  float* norm = carve((size_t)E * 4);
  float* xw   = carve((size_t)N * H1 * 4);   // xw1, later reused for xw2 (H2 <= H1)
  float* agg  = carve((size_t)N * H1 * 4);   // agg1/h1, later reused for agg2/h2
  (void)ws_size;

  const int T = 256;
  const long NH1 = (long)N * H1, NH2 = (long)N * H2;
  const int edgeBlocks = (int)(((long)E * 32 + T - 1) / T);

  auto scatter = [&](float* aggp, const float* xwp, int M) {
    if (M == 128)
      k_scatter_vec<4><<<edgeBlocks, T, 0, stream>>>(aggp, xwp, norm, src, dst, E);
    else if (M == 64)
      k_scatter_vec<2><<<edgeBlocks, T, 0, stream>>>(aggp, xwp, norm, src, dst, E);
    else
      k_scatter_generic<<<edgeBlocks, T, 0, stream>>>(aggp, xwp, norm, src, dst, E, M);
  };

  // degree + normalization
  k_init_deg     <<<(N + T - 1) / T, T, 0, stream>>>(dinv, N);
  k_accum_deg    <<<(E + T - 1) / T, T, 0, stream>>>(dinv, dst, ew, E);
  k_rsqrt_inplace<<<(N + T - 1) / T, T, 0, stream>>>(dinv, N);
  k_edge_norm    <<<(E + T - 1) / T, T, 0, stream>>>(norm, dinv, src, dst, ew, E);

  // layer 1: xw1 = x @ W1 ; agg1 = self + scatter ; h1 = relu(agg1 + b1)
  k_gemm_wmma_f32<<<dim3((N + 15) / 16), dim3(32, H1 / 16), 0, stream>>>(x, W1, xw, N, F, H1);
  k_self_init    <<<(int)((NH1 + T - 1) / T), T, 0, stream>>>(agg, xw, dinv, NH1, sh1);
  scatter(agg, xw, H1);
  k_bias_relu    <<<(int)((NH1 + T - 1) / T), T, 0, stream>>>(agg, b1, NH1, sh1);

  // layer 2: xw2 = h1 @ W2 ; agg2 ; h2 = relu(agg2 + b2)   (buffers reused, stream-ordered)
  k_gemm_wmma_f32<<<dim3((N + 15) / 16), dim3(32, H2 / 16), 0, stream>>>(agg, W2, xw, N, H1, H2);
  k_self_init    <<<(int)((NH2 + T - 1) / T), T, 0, stream>>>(agg, xw, dinv, NH2, sh2);
  scatter(agg, xw, H2);
  k_bias_relu    <<<(int)((NH2 + T - 1) / T), T, 0, stream>>>(agg, b2, NH2, sh2);

  // final projection
  k_fc<<<(N + T - 1) / T, T, 0, stream>>>(out, agg, Wfc, bfc, N, H2);
}